// KNN_65773129171296
// MI455X (gfx1250) — compile-verified
//
#include <hip/hip_runtime.h>

typedef __attribute__((ext_vector_type(2))) float v2f;
typedef __attribute__((ext_vector_type(8))) float v8f;

#define WAVES_PER_BLOCK 4
#define TILE 16
#define K_NN 16
#define LDS_STRIDE 20   // 16 + 4 pad floats -> 80B rows (16B aligned, conflict-free)

// Branchless sorted-list insert: chain of conditional swaps (v_cmp + v_cndmask).
__device__ __forceinline__ void insert16(float (&bd)[K_NN], int (&bi)[K_NN],
                                         float dd, int ii) {
#pragma unroll
  for (int j = 0; j < K_NN; ++j) {
    const bool  c  = dd < bd[j];
    const float fj = bd[j];
    const int   ij = bi[j];
    bd[j] = c ? dd : fj;
    bi[j] = c ? ii : ij;
    dd    = c ? fj : dd;
    ii    = c ? ij : ii;
  }
}

// Scan 8 row-half values: one min-tree reject for the whole group (common case),
// individual guarded inserts only when the group minimum beats the current worst.
__device__ __forceinline__ void scan8(float (&bd)[K_NN], int (&bi)[K_NN],
                                      const float4 q0, const float4 q1, int base) {
  const float m = fminf(fminf(fminf(q0.x, q0.y), fminf(q0.z, q0.w)),
                        fminf(fminf(q1.x, q1.y), fminf(q1.z, q1.w)));
  if (m < bd[K_NN - 1]) {
    const float vals[8] = {q0.x, q0.y, q0.z, q0.w, q1.x, q1.y, q1.z, q1.w};
#pragma unroll
    for (int c = 0; c < 8; ++c)
      if (vals[c] < bd[K_NN - 1]) insert16(bd, bi, vals[c], base + c);
  }
}

// One wave32 owns a 16-row seed tile. Per iteration, TWO candidate tiles of 16
// points each are evaluated with two independent V_WMMA_F32_16X16X4_F32:
//   A = [-2sx,-2sy,-2sz,1] (16x4),  B = [px,py,pz,|p|^2] (4x16)
//   => D[i][j] = |p_j|^2 - 2 s_i . p_j   (rank-equivalent to squared distance)
// Selection: lane r owns row r cols 0-7, lane r+16 owns row r cols 8-15; the
// two per-half top-16 lists are merged once at the end.
__global__ __launch_bounds__(WAVES_PER_BLOCK * 32) void knn_wmma_kernel(
    const float* __restrict__ seed,   // [B*N*3]
    const float* __restrict__ pts,    // [B*M*3]
    int* __restrict__ out,            // [B*N*K_NN]
    int N, int M)
{
  __shared__ float tbuf[WAVES_PER_BLOCK][2][2][TILE][LDS_STRIDE]; // [parity][tile]
  __shared__ float mergeD[WAVES_PER_BLOCK][32][K_NN + 1];  // +1 = +inf sentinel
  __shared__ int   mergeI[WAVES_PER_BLOCK][32][K_NN + 1];

  const int  lane = threadIdx.x & 31;
  const int  wave = threadIdx.x >> 5;
  const int  col  = lane & 15;
  const bool hi   = lane >= 16;

  const int rowTile = blockIdx.x * WAVES_PER_BLOCK + wave;
  const int rowBase = rowTile * TILE;              // flattened row in [0, B*N)
  const int batch   = rowBase / N;
  const float* bp   = pts + (size_t)batch * (size_t)M * 3;

  // ---- A matrix (fixed for the whole scan) ----
  // 16x4 f32 A layout: lanes 0-15 hold K=0/K=1; lanes 16-31 hold K=2/K=3.
  const float* srow = seed + (size_t)(rowBase + col) * 3;
  const float sx = srow[0], sy = srow[1], sz = srow[2];
  v2f A;
  A.x = hi ? (-2.0f * sz) : (-2.0f * sx);
  A.y = hi ? 1.0f         : (-2.0f * sy);

  // ---- per-lane top-k state (half a row each), sorted ascending ----
  float bestD[K_NN];
  int   bestI[K_NN];
#pragma unroll
  for (int j = 0; j < K_NN; ++j) { bestD[j] = __builtin_inff(); bestI[j] = 0; }

  const int numTiles = M / TILE;                   // 1024
  const int half = hi ? 8 : 0;
  for (int t = 0; t < numTiles; t += 2) {
    // Two candidate points per lane: tile t col, tile t+1 col.
    const float* p0 = bp + (size_t)((t    ) * TILE + col) * 3;
    const float* p1 = bp + (size_t)((t + 1) * TILE + col) * 3;
    const float ax = p0[0], ay = p0[1], az = p0[2];
    const float cx = p1[0], cy = p1[1], cz = p1[2];
    if (t + 9 < numTiles)   // low lanes prefetch tile t+8, high lanes tile t+9
      __builtin_prefetch(bp + (size_t)((t + 8 + (hi ? 1 : 0)) * TILE + col) * 3, 0, 0);

    // 4x16 f32 B layout mirrors A: lanes 0-15 -> K=0/K=1; lanes 16-31 -> K=2/K=3.
    v2f B0, B1;
    B0.x = hi ? az : ax;
    B0.y = hi ? (ax * ax + ay * ay + az * az) : ay;
    B1.x = hi ? cz : cx;
    B1.y = hi ? (cx * cx + cy * cy + cz * cz) : cy;

    v8f C = {};
    // Two independent WMMA chains per barrier interval.
    v8f D0 = __builtin_amdgcn_wmma_f32_16x16x4_f32(
        false, A, false, B0, (short)0, C, false, false);
    v8f D1 = __builtin_amdgcn_wmma_f32_16x16x4_f32(
        false, A, false, B1, (short)0, C, false, false);

    // ---- transpose both tiles through per-wave LDS ----
    // D: VGPR v, lanes 0-15 -> (M=v, N=lane); lanes 16-31 -> (M=v+8, N=lane-16)
    const int par = (t >> 1) & 1;
    float* tb0 = &tbuf[wave][par][0][0][0];
    float* tb1 = &tbuf[wave][par][1][0][0];
    const int rsel = hi ? 8 : 0;
#pragma unroll
    for (int v = 0; v < 8; ++v) {
      tb0[(v + rsel) * LDS_STRIDE + col] = D0[v];
      tb1[(v + rsel) * LDS_STRIDE + col] = D1[v];
    }

    __syncthreads();   // double-buffered: one barrier per pair of tiles

    // lane r: row r cols 0-7; lane r+16: row r cols 8-15 (16B aligned)
    const float4* r0 = (const float4*)&tb0[col * LDS_STRIDE + half];
    const float4* r1 = (const float4*)&tb1[col * LDS_STRIDE + half];
    const float4 a0 = r0[0], a1 = r0[1];
    const float4 b0 = r1[0], b1 = r1[1];
    scan8(bestD, bestI, a0, a1, (t    ) * TILE + half);
    scan8(bestD, bestI, b0, b1, (t + 1) * TILE + half);
  }

  // ---- epilogue: merge the two sorted half-row lists via LDS ----
#pragma unroll
  for (int j = 0; j < K_NN; ++j) {
    mergeD[wave][lane][j] = bestD[j];
    mergeI[wave][lane][j] = bestI[j];
  }
  mergeD[wave][lane][K_NN] = __builtin_inff();   // exhaustion sentinel
  mergeI[wave][lane][K_NN] = 0x7fffffff;
  __syncthreads();

  if (!hi) {
    const float* aD = mergeD[wave][lane];
    const int*   aI = mergeI[wave][lane];
    const float* bD = mergeD[wave][lane + 16];
    const int*   bI = mergeI[wave][lane + 16];
    int* orow = out + (size_t)(rowBase + lane) * K_NN;
    int ai = 0, bj = 0;
    for (int j = 0; j < K_NN; ++j) {
      const float da = aD[ai], db = bD[bj];
      const int   ia = aI[ai], ib = bI[bj];
      // stable tie-break: smaller index wins on equal distance (matches top_k)
      const bool takeA = (da < db) || ((da == db) && (ia < ib));
      orow[j] = takeA ? ia : ib;
      ai += takeA ? 1 : 0;
      bj += takeA ? 0 : 1;
    }
  }
}

extern "C" void kernel_launch(void* const* d_in, const int* in_sizes, int n_in,
                              void* d_out, int out_size, void* d_ws, size_t ws_size,
                              hipStream_t stream) {
  const float* seed = (const float*)d_in[0];  // [2,16384,3] f32
  const float* pts  = (const float*)d_in[1];  // [2,16384,3] f32
  int* out = (int*)d_out;                     // [2,16384,16] int32 indices

  const int totalRows = in_sizes[0] / 3;      // B*N = 32768
  const int N = 16384;
  const int M = 16384;

  const int blocks = totalRows / (WAVES_PER_BLOCK * TILE); // 512
  knn_wmma_kernel<<<blocks, WAVES_PER_BLOCK * 32, 0, stream>>>(seed, pts, out, N, M);
}